// GATLayer_57062935494774
// MI455X (gfx1250) — compile-verified
//
#include <hip/hip_runtime.h>
#include <math.h>

#define V_NODES 50000
#define E_EDGES 800000
#define IN_DIM  256
#define OUT_DIM 128
#define ALPHA   0.2f

typedef __attribute__((ext_vector_type(2))) float v2f;
typedef __attribute__((ext_vector_type(8))) float v8f;

// Pointer/vector types for the async-to-LDS path. The builtin takes
// (v4i addrspace(1)* src, v4i addrspace(3)* dst, imm offset, imm cpol).
typedef int v4i __attribute__((vector_size(16)));
typedef __attribute__((address_space(1))) v4i gas_v4i;
typedef __attribute__((address_space(3))) v4i las_v4i;
typedef __attribute__((address_space(3))) char las_char;

#if defined(__has_builtin)
#if __has_builtin(__builtin_amdgcn_global_load_async_to_lds_b128)
#define HAVE_ASYNC_B128 1
#endif
#if __has_builtin(__builtin_amdgcn_s_wait_asynccnt)
#define HAVE_WAIT_ASYNC 1
#endif
#endif

// ---------------------------------------------------------------------------
// Kernel 1: Wh = x @ W using V_WMMA_F32_16X16X4_F32 (full fp32 accuracy).
// Grid: 3125 blocks (M tiles of 16 rows). Block: 256 threads = 8 wave32s.
// Each wave computes one 16x16 output tile (8 N-tiles cover OUT=128).
// A-strip (16x256 f32) copied global->LDS with GLOBAL_LOAD_ASYNC_TO_LDS_B128
// (ASYNCcnt path), +4 float row padding -> bank-conflict-free LDS reads.
// ---------------------------------------------------------------------------
__global__ __launch_bounds__(256) void gat_gemm_wmma(const float* __restrict__ x,
                                                     const float* __restrict__ W,
                                                     float* __restrict__ Wh) {
  __shared__ float As[16][IN_DIM + 4]; // row stride 260 floats (1040B, 16B aligned)
  const int m0  = blockIdx.x * 16;
  const int tid = threadIdx.x;

  // Async copy of the 16x256 A strip into LDS: 1024 x 16B, 4 per thread.
  las_char* lbase = (las_char*)(void*)&As[0][0];
#pragma unroll
  for (int i = 0; i < 4; ++i) {
    int idx = tid + 256 * i;          // 0..1023
    int row = idx >> 6;               // / 64 float4s per row
    int c4  = idx & 63;
    const float* gsrc = x + (size_t)(m0 + row) * IN_DIM + c4 * 4;
    unsigned loff = (unsigned)(row * (IN_DIM + 4) + c4 * 4) * 4u;
#if defined(HAVE_ASYNC_B128)
    __builtin_amdgcn_global_load_async_to_lds_b128(
        (gas_v4i*)(void*)(const void*)gsrc,
        (las_v4i*)(void*)(lbase + loff), 0, 0);
#else
    unsigned lds_addr = (unsigned)(uintptr_t)lbase + loff;
    unsigned long long gaddr = (unsigned long long)(uintptr_t)gsrc;
    asm volatile("global_load_async_to_lds_b128 %0, %1, off"
                 :: "v"(lds_addr), "v"(gaddr) : "memory");
#endif
  }
#if defined(HAVE_WAIT_ASYNC)
  __builtin_amdgcn_s_wait_asynccnt(0);
#else
  asm volatile("s_wait_asynccnt 0x0" ::: "memory");
#endif
  __syncthreads();

  const int wave = tid >> 5;
  const int lane = tid & 31;
  const int n0   = wave * 16;         // this wave's N tile
  const int mrow = lane & 15;         // A: lane -> M row
  const int kb   = (lane >> 4) * 2;   // A/B: upper half-lanes hold K+2,K+3
  const int ncol = lane & 15;         // B/D: lane -> N col

  v8f acc = {};
#pragma unroll 4
  for (int k = 0; k < IN_DIM; k += 4) {
    v2f a;
    a.x = As[mrow][k + kb];
    a.y = As[mrow][k + kb + 1];
    v2f b;
    b.x = W[(size_t)(k + kb) * OUT_DIM + n0 + ncol];
    b.y = W[(size_t)(k + kb + 1) * OUT_DIM + n0 + ncol];
    acc = __builtin_amdgcn_wmma_f32_16x16x4_f32(false, a, false, b, (short)0, acc,
                                                false, false);
  }

  // D layout: VGPR r, lanes 0-15 -> M=r, lanes 16-31 -> M=r+8; N=lane&15.
  const int mhi = (lane >> 4) * 8;
#pragma unroll
  for (int r = 0; r < 8; ++r) {
    Wh[(size_t)(m0 + r + mhi) * OUT_DIM + n0 + ncol] = acc[r];
  }
}

// ---------------------------------------------------------------------------
// Kernel 2: s_src[v] = Wh[v].a[:OUT], s_dst[v] = Wh[v].a[OUT:]. One wave/node.
// ---------------------------------------------------------------------------
__global__ __launch_bounds__(256) void gat_scores(const float* __restrict__ Wh,
                                                  const float* __restrict__ a,
                                                  float* __restrict__ s_src,
                                                  float* __restrict__ s_dst) {
  const int v    = blockIdx.x * 8 + (threadIdx.x >> 5); // 6250*8 == 50000 exact
  const int lane = threadIdx.x & 31;
  const float* whv = Wh + (size_t)v * OUT_DIM;
  float ss = 0.f, sd = 0.f;
#pragma unroll
  for (int j = 0; j < 4; ++j) {
    float w = whv[lane + 32 * j];
    ss += w * a[lane + 32 * j];
    sd += w * a[OUT_DIM + lane + 32 * j];
  }
#pragma unroll
  for (int off = 16; off > 0; off >>= 1) {
    ss += __shfl_xor(ss, off);
    sd += __shfl_xor(sd, off);
  }
  if (lane == 0) {
    s_src[v] = ss;
    s_dst[v] = sd;
  }
}

// ---------------------------------------------------------------------------
// Kernel 3: zero the degree histogram.
// ---------------------------------------------------------------------------
__global__ void gat_zero_cnt(int* __restrict__ cnt, int n) {
  int i = blockIdx.x * blockDim.x + threadIdx.x;
  if (i < n) cnt[i] = 0;
}

// ---------------------------------------------------------------------------
// Kernel 4: per-edge degree histogram.
// ---------------------------------------------------------------------------
__global__ void gat_count(const int* __restrict__ row, int* __restrict__ cnt) {
  int e = blockIdx.x * blockDim.x + threadIdx.x;
  if (e < E_EDGES) atomicAdd(&cnt[row[e]], 1);
}

// ---------------------------------------------------------------------------
// Kernel 5: single-workgroup exclusive scan -> offsets[V+1] and cursor[V].
// ---------------------------------------------------------------------------
__global__ __launch_bounds__(1024) void gat_scan(const int* __restrict__ cnt,
                                                 int* __restrict__ offsets,
                                                 int* __restrict__ cursor) {
  __shared__ int s[1024];
  __shared__ int carry;
  if (threadIdx.x == 0) carry = 0;
  __syncthreads();
  for (int base = 0; base < V_NODES; base += 1024) {
    int i = base + (int)threadIdx.x;
    int v = (i < V_NODES) ? cnt[i] : 0;
    s[threadIdx.x] = v;
    __syncthreads();
    // Hillis-Steele inclusive scan.
    for (int off = 1; off < 1024; off <<= 1) {
      int t = (threadIdx.x >= (unsigned)off) ? s[threadIdx.x - off] : 0;
      __syncthreads();
      s[threadIdx.x] += t;
      __syncthreads();
    }
    int excl = carry + s[threadIdx.x] - v;
    if (i < V_NODES) {
      offsets[i] = excl;
      cursor[i]  = excl;
    }
    __syncthreads();
    if (threadIdx.x == 1023) carry += s[1023];
    __syncthreads();
  }
  if (threadIdx.x == 0) offsets[V_NODES] = carry; // == E
}

// ---------------------------------------------------------------------------
// Kernel 6: scatter edges into CSR, precomputing LeakyReLU scores.
// ---------------------------------------------------------------------------
__global__ void gat_scatter(const int* __restrict__ row, const int* __restrict__ col,
                            const float* __restrict__ s_src,
                            const float* __restrict__ s_dst,
                            int* __restrict__ cursor,
                            int* __restrict__ col_s, float* __restrict__ e_s) {
  int e = blockIdx.x * blockDim.x + threadIdx.x;
  if (e >= E_EDGES) return;
  int r = row[e], c = col[e];
  float sc = s_src[r] + s_dst[c];
  float ev = sc > 0.f ? sc : ALPHA * sc;
  int p = atomicAdd(&cursor[r], 1);
  col_s[p] = c;
  e_s[p]   = ev;
}

// ---------------------------------------------------------------------------
// Kernel 7: per-row softmax + weighted aggregation + ELU. One wave32 per row;
// 128 channels split 4-per-lane; Wh gathers are L2-resident (25.6MB << 192MB).
// Next edge's Wh row is prefetched (global_prefetch_b8) to hide L2 latency.
// ---------------------------------------------------------------------------
__global__ __launch_bounds__(256) void gat_aggregate(const int* __restrict__ offsets,
                                                     const int* __restrict__ col_s,
                                                     const float* __restrict__ e_s,
                                                     const float* __restrict__ Wh,
                                                     float* __restrict__ out) {
  const int r    = blockIdx.x * 8 + (threadIdx.x >> 5); // 6250*8 == 50000 exact
  const int lane = threadIdx.x & 31;
  const int beg = offsets[r];
  const int end = offsets[r + 1];

  // segment max
  float m = -INFINITY;
  for (int i = beg + lane; i < end; i += 32) m = fmaxf(m, e_s[i]);
#pragma unroll
  for (int off = 16; off > 0; off >>= 1) m = fmaxf(m, __shfl_xor(m, off));

  // segment sum of exp
  float den = 0.f;
  for (int i = beg + lane; i < end; i += 32) den += __expf(e_s[i] - m);
#pragma unroll
  for (int off = 16; off > 0; off >>= 1) den += __shfl_xor(den, off);
  float inv = (end > beg) ? 1.0f / den : 0.0f;

  // weighted aggregation: serial over ~deg edges, all lanes cover 128 channels
  float a0 = 0.f, a1 = 0.f, a2 = 0.f, a3 = 0.f;
  for (int i = beg; i < end; ++i) {
    if (i + 1 < end) {
      // lanes spread over the 4 cachelines of the next 512B row
      __builtin_prefetch(Wh + (size_t)col_s[i + 1] * OUT_DIM + lane * 4, 0, 0);
    }
    float w = __expf(e_s[i] - m) * inv;
    const float* whc = Wh + (size_t)col_s[i] * OUT_DIM;
    a0 += w * whc[lane];
    a1 += w * whc[lane + 32];
    a2 += w * whc[lane + 64];
    a3 += w * whc[lane + 96];
  }

  float* o = out + (size_t)r * OUT_DIM;
  o[lane]      = a0 > 0.f ? a0 : __expf(a0) - 1.0f;
  o[lane + 32] = a1 > 0.f ? a1 : __expf(a1) - 1.0f;
  o[lane + 64] = a2 > 0.f ? a2 : __expf(a2) - 1.0f;
  o[lane + 96] = a3 > 0.f ? a3 : __expf(a3) - 1.0f;
}

// ---------------------------------------------------------------------------
extern "C" void kernel_launch(void* const* d_in, const int* in_sizes, int n_in,
                              void* d_out, int out_size, void* d_ws, size_t ws_size,
                              hipStream_t stream) {
  const float* x  = (const float*)d_in[0];          // (V, 256)
  const float* W  = (const float*)d_in[1];          // (256, 128)
  const float* a  = (const float*)d_in[2];          // (256,)
  const int*   ei = (const int*)d_in[3];            // (2, E)
  const int* row = ei;
  const int* col = ei + E_EDGES;
  float* out = (float*)d_out;                       // (V, 128)

  // Workspace layout (all 4-byte typed; ~33 MB total).
  float* Wh      = (float*)d_ws;                    // V*128
  float* s_src   = Wh + (size_t)V_NODES * OUT_DIM;  // V
  float* s_dst   = s_src + V_NODES;                 // V
  int*   cnt     = (int*)(s_dst + V_NODES);         // V
  int*   offsets = cnt + V_NODES;                   // V+1
  int*   cursor  = offsets + V_NODES + 1;           // V
  int*   col_s   = cursor + V_NODES;                // E
  float* e_s     = (float*)(col_s + E_EDGES);       // E

  // 1. Wh = x @ W  (WMMA f32, async-to-LDS staging)
  gat_gemm_wmma<<<V_NODES / 16, 256, 0, stream>>>(x, W, Wh);
  // 2. attention score dots
  gat_scores<<<V_NODES / 8, 256, 0, stream>>>(Wh, a, s_src, s_dst);
  // 3-5. build CSR
  gat_zero_cnt<<<(V_NODES + 255) / 256, 256, 0, stream>>>(cnt, V_NODES);
  gat_count<<<(E_EDGES + 255) / 256, 256, 0, stream>>>(row, cnt);
  gat_scan<<<1, 1024, 0, stream>>>(cnt, offsets, cursor);
  gat_scatter<<<(E_EDGES + 255) / 256, 256, 0, stream>>>(row, col, s_src, s_dst,
                                                         cursor, col_s, e_s);
  // 7. softmax + aggregate + ELU
  gat_aggregate<<<V_NODES / 8, 256, 0, stream>>>(offsets, col_s, e_s, Wh, out);
}